// KGATEnhanced_67654324846923
// MI455X (gfx1250) — compile-verified
//
#include <hip/hip_runtime.h>

#define D     64
#define NNEI  50
#define ROWS  16
#define BLK   128   // 4 waves of 32

typedef float v2f __attribute__((ext_vector_type(2)));
typedef float v8f __attribute__((ext_vector_type(8)));

// One wave computes one 16x16 N-tile of C = A(16x64, LDS) * W(64x64 global, row-major),
// accumulating into c. A-tile layout (f32 16x4): lane hl = M, vgpr0/1 = K(2*half)+0/1.
// B-tile layout (f32 4x16): vgpr0 = row K(2*half), vgpr1 = row K(2*half+1), col = hl.
__device__ __forceinline__ void wmma_k64_tile(v8f& c, const float* __restrict__ A_lds,
                                              const float* __restrict__ W,
                                              int nt, int hl, int half) {
  for (int kc = 0; kc < 16; ++kc) {
    const int k0 = 4 * kc + 2 * half;
    v2f a;
    a.x = A_lds[hl * D + k0];
    a.y = A_lds[hl * D + k0 + 1];
    v2f b;
    b.x = W[(size_t)k0 * D + nt * 16 + hl];
    b.y = W[(size_t)(k0 + 1) * D + nt * 16 + hl];
    c = __builtin_amdgcn_wmma_f32_16x16x4_f32(
        /*neg_a=*/false, a, /*neg_b=*/false, b,
        /*c_mod=*/(short)0, c, /*reuse_a=*/false, /*reuse_b=*/false);
  }
}

// C/D layout (f32 16x16): vgpr i -> row M = i + 8*half, col N = hl (+ nt*16)
__device__ __forceinline__ void wmma_store_tile(const v8f& c, const float* __restrict__ bias,
                                                bool relu, float* __restrict__ out_lds,
                                                int nt, int hl, int half) {
  const int col = nt * 16 + hl;
  const float bv = bias[col];
#pragma unroll
  for (int i = 0; i < 8; ++i) {
    const int row = i + 8 * half;
    float v = c[i] + bv;
    if (relu) v = fmaxf(v, 0.f);
    out_lds[row * D + col] = v;
  }
}

__global__ __launch_bounds__(BLK) void kgat_fused(
    const int* __restrict__ user_idx, const int* __restrict__ item_idx,
    const int* __restrict__ kg_adj,
    const float* __restrict__ user_table, const float* __restrict__ item_table,
    const float* __restrict__ entity_table,
    const float* __restrict__ attn_W, const float* __restrict__ attn_b,
    const float* __restrict__ kg_W1, const float* __restrict__ kg_b1,
    const float* __restrict__ kg_W2, const float* __restrict__ kg_b2,
    const float* __restrict__ comb_W, const float* __restrict__ comb_b,
    const float* __restrict__ out_W, const float* __restrict__ out_b,
    float* __restrict__ out, int B) {
  __shared__ __align__(16) float item_s[ROWS * D];
  __shared__ __align__(16) float bufA[ROWS * D];
  __shared__ __align__(16) float bufB[ROWS * D];
  __shared__ float scores_s[ROWS * NNEI];
  __shared__ float iscore_s[ROWS];
  __shared__ int nbr_s[ROWS * NNEI];
  __shared__ int iidx_s[ROWS];
  __shared__ int uidx_s[ROWS];

  const int tid = threadIdx.x;
  const int lane = tid & 31;
  const int wid = tid >> 5;           // wave id 0..3 == WMMA N-tile id
  const int hl = lane & 15, half = (lane >> 4) & 1;
  const int r0 = blockIdx.x * ROWS;

  // --- stage indices (clamped; B is a multiple of 16 in practice) ---
  if (tid < ROWS) {
    const int rr = min(r0 + tid, B - 1);
    iidx_s[tid] = item_idx[rr];
    uidx_s[tid] = user_idx[rr];
  }
  for (int t = tid; t < ROWS * NNEI; t += BLK) {
    const int rr = min(r0 + t / NNEI, B - 1);
    const int e = kg_adj[(size_t)rr * NNEI + (t % NNEI)];
    nbr_s[t] = e < 0 ? 0 : e;  // jnp.clip(idx, 0)
  }
  __syncthreads();

  // --- gather item embedding tile (16x64 = 256 float4, 2 per thread) ---
  for (int t = tid; t < ROWS * (D / 4); t += BLK) {
    const int row = t >> 4, c4 = t & 15;
    const float4* src = (const float4*)(item_table + (size_t)iidx_s[row] * D);
    ((float4*)item_s)[t] = src[c4];
  }
  __syncthreads();

  // --- per-row item score: dot(item_emb[r], Wi) + attn_b (16 threads, no WMMA here) ---
  if (tid < ROWS) {
    const float4* ip = (const float4*)(item_s + tid * D);
    const float4* wi = (const float4*)(attn_W);
    float s = 0.f;
#pragma unroll
    for (int j = 0; j < 16; ++j) {
      const float4 e = ip[j], w = wi[j];
      s += e.x * w.x + e.y * w.y + e.z * w.z + e.w * w.w;
    }
    iscore_s[tid] = s + attn_b[0];
  }
  __syncthreads();

  // --- pass 1: 800 attention scores over 128 threads (full 64-dot each, b128 loads) ---
  {
    float4 wn4[16];
    const float4* wn = (const float4*)(attn_W + D);
#pragma unroll
    for (int j = 0; j < 16; ++j) wn4[j] = wn[j];
    for (int t = tid; t < ROWS * NNEI; t += BLK) {
      const int row = t / NNEI;
      const int eidx = nbr_s[t];
      const float4* ep = (const float4*)(entity_table + (size_t)eidx * D);
      float s = 0.f;
#pragma unroll
      for (int j = 0; j < 16; ++j) {
        const float4 e = ep[j];
        s += e.x * wn4[j].x + e.y * wn4[j].y + e.z * wn4[j].z + e.w * wn4[j].w;
      }
      float sc = s + iscore_s[row];
      sc = sc > 0.f ? sc : 0.2f * sc;  // leaky relu, slope 0.2
      scores_s[t] = sc;
    }
  }
  __syncthreads();

  // --- softmax over 50 neighbors (16 threads, one row each) ---
  if (tid < ROWS) {
    float m = -3.402823466e38f;
    for (int n = 0; n < NNEI; ++n) m = fmaxf(m, scores_s[tid * NNEI + n]);
    float sum = 0.f;
    for (int n = 0; n < NNEI; ++n) {
      const float e = __expf(scores_s[tid * NNEI + n] - m);
      scores_s[tid * NNEI + n] = e;
      sum += e;
    }
    const float inv = 1.f / sum;
    for (int n = 0; n < NNEI; ++n) scores_s[tid * NNEI + n] *= inv;
  }
  __syncthreads();

  // --- pass 2: weighted neighbor sum; thread = (row, 8-col chunk); L2-resident re-gather ---
  {
    const int row = tid >> 3;        // 0..15
    const int j0 = tid & 7;          // 8-float chunk -> float4 index 2*j0, 2*j0+1
    float4 a0 = make_float4(0.f, 0.f, 0.f, 0.f);
    float4 a1 = a0;
    for (int n = 0; n < NNEI; ++n) {
      const float w = scores_s[row * NNEI + n];
      const int eidx = nbr_s[row * NNEI + n];
      const float4* ep = (const float4*)(entity_table + (size_t)eidx * D) + 2 * j0;
      const float4 e0 = ep[0], e1 = ep[1];
      a0.x += w * e0.x; a0.y += w * e0.y; a0.z += w * e0.z; a0.w += w * e0.w;
      a1.x += w * e1.x; a1.y += w * e1.y; a1.z += w * e1.z; a1.w += w * e1.w;
    }
    float4* op = (float4*)(bufA + row * D) + 2 * j0;
    op[0] = a0;
    op[1] = a1;
  }
  __syncthreads();

  v8f c;

  // --- GEMM1: bufB = relu(bufA @ kg_W1 + b1); wave w owns N-tile w ---
  c = (v8f){0, 0, 0, 0, 0, 0, 0, 0};
  wmma_k64_tile(c, bufA, kg_W1, wid, hl, half);
  wmma_store_tile(c, kg_b1, true, bufB, wid, hl, half);
  __syncthreads();

  // --- GEMM2: bufA = bufB @ kg_W2 + b2 (refined_kg) ---
  c = (v8f){0, 0, 0, 0, 0, 0, 0, 0};
  wmma_k64_tile(c, bufB, kg_W2, wid, hl, half);
  wmma_store_tile(c, kg_b2, false, bufA, wid, hl, half);
  __syncthreads();

  // --- GEMM3 (K=128): bufB = relu([item_s | bufA] @ comb_W + comb_b) ---
  c = (v8f){0, 0, 0, 0, 0, 0, 0, 0};
  wmma_k64_tile(c, item_s, comb_W, wid, hl, half);        // rows 0..63 of comb_W
  wmma_k64_tile(c, bufA, comb_W + D * D, wid, hl, half);  // rows 64..127
  wmma_store_tile(c, comb_b, true, bufB, wid, hl, half);
  __syncthreads();

  // --- final score: dot([user_emb | final_rep], out_W) + out_b; 8 lanes per row ---
  {
    const int row = tid >> 3;        // 0..15
    const int j0 = tid & 7;
    const int uidx = uidx_s[row];
    const float4* up  = (const float4*)(user_table + (size_t)uidx * D) + 2 * j0;
    const float4* w0p = (const float4*)(out_W) + 2 * j0;
    const float4* w1p = (const float4*)(out_W + D) + 2 * j0;
    const float4* fp  = (const float4*)(bufB + row * D) + 2 * j0;
    float s = 0.f;
#pragma unroll
    for (int j = 0; j < 2; ++j) {
      const float4 u = up[j], w0 = w0p[j];
      const float4 f = fp[j], w1 = w1p[j];
      s += u.x * w0.x + u.y * w0.y + u.z * w0.z + u.w * w0.w;
      s += f.x * w1.x + f.y * w1.y + f.z * w1.z + f.w * w1.w;
    }
    // reduce across the 8 lanes of this row-group (lanes are contiguous within a wave)
    s += __shfl_xor(s, 1, 32);
    s += __shfl_xor(s, 2, 32);
    s += __shfl_xor(s, 4, 32);
    if (j0 == 0 && (r0 + row) < B) out[r0 + row] = s + out_b[0];
  }
}

extern "C" void kernel_launch(void* const* d_in, const int* in_sizes, int n_in,
                              void* d_out, int out_size, void* d_ws, size_t ws_size,
                              hipStream_t stream) {
  const int*   user_idx     = (const int*)d_in[0];
  const int*   item_idx     = (const int*)d_in[1];
  const int*   kg_adj       = (const int*)d_in[2];
  const float* user_table   = (const float*)d_in[3];
  const float* item_table   = (const float*)d_in[4];
  const float* entity_table = (const float*)d_in[5];
  const float* attn_W = (const float*)d_in[6];
  const float* attn_b = (const float*)d_in[7];
  const float* kg_W1  = (const float*)d_in[8];
  const float* kg_b1  = (const float*)d_in[9];
  const float* kg_W2  = (const float*)d_in[10];
  const float* kg_b2  = (const float*)d_in[11];
  const float* comb_W = (const float*)d_in[12];
  const float* comb_b = (const float*)d_in[13];
  const float* out_W  = (const float*)d_in[14];
  const float* out_b  = (const float*)d_in[15];
  float* out = (float*)d_out;

  const int B = in_sizes[0];
  dim3 grid((B + ROWS - 1) / ROWS), block(BLK);
  kgat_fused<<<grid, block, 0, stream>>>(
      user_idx, item_idx, kg_adj, user_table, item_table, entity_table,
      attn_W, attn_b, kg_W1, kg_b1, kg_W2, kg_b2, comb_W, comb_b,
      out_W, out_b, out, B);
}